// ErdosLoss_26843545600697
// MI455X (gfx1250) — compile-verified
//
#include <hip/hip_runtime.h>
#include <hip/hip_bf16.h>

#define N_NODES 4000
#define N_EDGES 6000
#define PENALTY_SCALE (16.0f * 200.0f * 3.0f)

typedef __attribute__((ext_vector_type(2))) float v2f;
typedef __attribute__((ext_vector_type(8))) float v8f;

// Wave-wide sum of `v` across all 32 lanes using the matrix pipe:
//   D = A x B + 0 with B = all-ones (row-sum regardless of B's lane layout),
//   A holds one value per lane (other K slot zeroed).
// Codegen-verified C/D layout: lane n<16 holds D[0..7][n], lane n>=16 holds
// D[8..15][n-16]. So (sum of 8 regs in lane 0) + (sum in lane 16) == wave sum.
// Those two lanes commit via global_atomic_add_f32 -> no shuffle chain needed.
// Requires EXEC all-ones (call from fully-populated, reconverged blocks only).
__device__ __forceinline__ void wave_sum_wmma_atomic(float v, float* dst) {
    v2f a; a.x = v;    a.y = 0.0f;
    v2f b; b.x = 1.0f; b.y = 1.0f;
    v8f c = {};
    // 8 args: (neg_a, A, neg_b, B, c_mod, C, reuse_a, reuse_b)
    c = __builtin_amdgcn_wmma_f32_16x16x4_f32(
        /*neg_a=*/false, a, /*neg_b=*/false, b,
        /*c_mod=*/(short)0, c, /*reuse_a=*/false, /*reuse_b=*/false);
    float partial = ((c[0] + c[1]) + (c[2] + c[3])) +
                    ((c[4] + c[5]) + (c[6] + c[7]));
    unsigned lane = threadIdx.x & 31u;   // wave32
    if (lane == 0u || lane == 16u) atomicAdd(dst, partial);
}

// Zero the workspace accumulators (logS[N], S[N], acc[4]).
__global__ void k_zero(float* __restrict__ ws, int n) {
    int i = blockIdx.x * blockDim.x + threadIdx.x;
    if (i < n) ws[i] = 0.0f;
}

// Per-edge scatter:
//   logS[dst] += log(1 - p + 1e-6)
//   S[src]    += p ;  S[dst] += p (once only for self-loops)
//   diag      += p^2 * (1 if self-loop else 2)   -> acc[2]
__global__ void k_edges(const int* __restrict__ ei, const float* __restrict__ ef,
                        float* __restrict__ logS, float* __restrict__ S,
                        float* __restrict__ acc) {
    int e = blockIdx.x * blockDim.x + threadIdx.x;
    float diag = 0.0f;
    if (e < N_EDGES) {
        int   src = ei[e];
        int   dst = ei[N_EDGES + e];
        float p   = ef[e];
        atomicAdd(&logS[dst], logf(1.0f - p + 1e-6f));
        atomicAdd(&S[src], p);
        if (src != dst) atomicAdd(&S[dst], p);
        diag = p * p * ((src == dst) ? 1.0f : 2.0f);
    }
    // Reconverged; full blocks (24x256 covers 6000) -> EXEC all ones here.
    wave_sum_wmma_atomic(diag, &acc[2]);
}

// Per-node terms:
//   acc[0] += exp(logS[n])   (empty segments -> exp(0)=1, matches segment_sum)
//   acc[1] += S[n]^2
__global__ void k_nodes(const float* __restrict__ logS, const float* __restrict__ S,
                        float* __restrict__ acc) {
    int n = blockIdx.x * blockDim.x + threadIdx.x;
    float te = 0.0f, ts = 0.0f;
    if (n < N_NODES) {
        te = expf(logS[n]);
        float s = S[n];
        ts = s * s;
    }
    wave_sum_wmma_atomic(te, &acc[0]);
    wave_sum_wmma_atomic(ts, &acc[1]);
}

// out = SCALE * acc0/N  +  200 * 0.5*(acc1 - acc2) / (batch[N-1] + 1)
// (batch is sorted ascending -> max is the last element)
__global__ void k_final(const float* __restrict__ acc, const int* __restrict__ batch,
                        float* __restrict__ out) {
    float num_graphs = (float)(batch[N_NODES - 1] + 1);
    float loss2 = acc[0] * (PENALTY_SCALE / (float)N_NODES);
    float loss3 = 0.5f * (acc[1] - acc[2]);
    out[0] = loss2 + 200.0f * loss3 / num_graphs;
}

extern "C" void kernel_launch(void* const* d_in, const int* in_sizes, int n_in,
                              void* d_out, int out_size, void* d_ws, size_t ws_size,
                              hipStream_t stream) {
    // inputs (setup_inputs order): x[4000] f32 (unused), edge_index[2*6000] i32,
    // edge_feature[6000] f32, batch[4000] i32
    const int*   ei    = (const int*)d_in[1];
    const float* ef    = (const float*)d_in[2];
    const int*   batch = (const int*)d_in[3];
    float*       out   = (float*)d_out;

    float* ws   = (float*)d_ws;
    float* logS = ws;                 // [N_NODES]
    float* S    = ws + N_NODES;       // [N_NODES]
    float* acc  = ws + 2 * N_NODES;   // [4]: sum_exp, sum_S2, diag, pad
    const int nz = 2 * N_NODES + 4;

    k_zero <<<(nz + 255) / 256,       256, 0, stream>>>(ws, nz);
    k_edges<<<(N_EDGES + 255) / 256,  256, 0, stream>>>(ei, ef, logS, S, acc);
    k_nodes<<<(N_NODES + 255) / 256,  256, 0, stream>>>(logS, S, acc);
    k_final<<<1, 1, 0, stream>>>(acc, batch, out);
}